// GraphAttentionCell_7146825580625
// MI455X (gfx1250) — compile-verified
//
#include <hip/hip_runtime.h>
#include <hip/hip_bf16.h>

typedef __attribute__((ext_vector_type(2))) float v2f;
typedef __attribute__((ext_vector_type(8))) float v8f;
typedef __attribute__((ext_vector_type(4))) unsigned int u32x4;
typedef __attribute__((ext_vector_type(8))) int i32x8;
typedef __attribute__((ext_vector_type(4))) int i32x4;

static constexpr int B_  = 2;
static constexpr int N_  = 1024;
static constexpr int F_  = 128;
static constexpr int C_  = 128;
static constexpr int ROWS = B_ * N_;   // 2048
static constexpr int OC   = 2 * C_;    // 256 (output feature dim)
static constexpr int JT   = 128;       // neighbor-chunk rows staged in LDS

// Full-precision CDNA5 matrix op: D(16x16,f32) = A(16x4,f32) x B(4x16,f32) + C
__device__ __forceinline__ v8f wmma_f32_k4(v2f a, v2f b, v8f c) {
  return __builtin_amdgcn_wmma_f32_16x16x4_f32(false, a, false, b, (short)0, c,
                                               false, false);
}

// ---------------------------------------------------------------------------
// TDM: async 2-D tile load (rows x C_ f32, row-major, stride C_) -> LDS.
// Descriptor per CDNA5 ISA ch.10.8 (D# group0 + group1); 2-D tensor so
// groups 2/3 are null. Tracked on TENSORcnt.
// ---------------------------------------------------------------------------
#if __has_builtin(__builtin_amdgcn_tensor_load_to_lds)
#define GAT_HAVE_TDM 1
__device__ __forceinline__ void tdm_load_rows(unsigned int lds_byte_off,
                                              unsigned long long gaddr,
                                              unsigned int rows) {
  u32x4 g0;
  g0[0] = 1u;                                             // count=1 (valid D#)
  g0[1] = lds_byte_off;                                   // lds_addr (bytes)
  g0[2] = (unsigned int)(gaddr & 0xFFFFFFFFull);          // global_addr[31:0]
  g0[3] = (unsigned int)((gaddr >> 32) & 0x01FFFFFFull)   // global_addr[56:32]
        | (2u << 30);                                     // type = 2 ("image")
  i32x8 g1;
  g1[0] = (int)(2u << 16);                // data_size=2 -> 4-byte elements
  g1[1] = (int)((unsigned)C_ << 16);      // tensor_dim0[15:0]  (bits 63:48)
  g1[2] = (int)(rows << 16);              // tensor_dim1[15:0]  (bits 95:80)
  g1[3] = (int)((unsigned)C_ << 16);      // tile_dim0          (bits 127:112)
  g1[4] = (int)rows;                      // tile_dim1          (bits 143:128)
  g1[5] = (int)C_;                        // tensor_dim0_stride (bits 191:160)
  g1[6] = 0;
  g1[7] = 0;
  i32x4 z = {0, 0, 0, 0};                 // groups 2/3: unused (2-D tensor)
#if defined(__clang_major__) && __clang_major__ >= 23
  i32x8 z8 = {0, 0, 0, 0, 0, 0, 0, 0};
  __builtin_amdgcn_tensor_load_to_lds(g0, g1, z, z, z8, 0);
#else
  __builtin_amdgcn_tensor_load_to_lds(g0, g1, z, z, 0);
#endif
}
#else
#define GAT_HAVE_TDM 0
#endif

// ---------------------------------------------------------------------------
// Kernel 1: feats = X @ W   ([2048 x 128] x [128 x 128])
// One wave owns a 16x16 output tile; K-loop = 32 x V_WMMA_F32_16X16X4_F32.
// Fused: out[:, :, 0:C] = relu(feats); raw feats -> workspace.
// ---------------------------------------------------------------------------
__global__ void __launch_bounds__(128)
gat_feats_gemm(const float* __restrict__ X, const float* __restrict__ W,
               float* __restrict__ feats, float* __restrict__ out) {
  const int tid  = threadIdx.x;
  const int wid  = blockIdx.x * 4 + (tid >> 5);   // 1024 tiles total
  const int lane = tid & 31;
  const int tile_m = wid >> 3;                    // 0..127 (rows of 16)
  const int tile_c = wid & 7;                     // 0..7   (cols of 16)
  const int r0 = tile_m * 16, c0 = tile_c * 16;
  const int r    = lane & 15;                     // row (A) / col (B,D) in tile
  const int half = lane >> 4;                     // K sub-slot selector

  const float* xrow = X + (size_t)(r0 + r) * F_;  // A-operand row
  const int col = c0 + r;                         // B/D column

  v8f acc = {};
#pragma unroll 4
  for (int k = 0; k < F_; k += 4) {
    const int ks = k + half * 2;
    v2f a; a.x = xrow[ks];             a.y = xrow[ks + 1];
    v2f b; b.x = W[ks * C_ + col];     b.y = W[(ks + 1) * C_ + col];
    acc = wmma_f32_k4(a, b, acc);
  }

#pragma unroll
  for (int v = 0; v < 8; ++v) {                   // D: VGPR v -> M = v + 8*half
    const int gm = r0 + v + 8 * half;             // global row in [0, 2048)
    const float f = acc[v];
    feats[(size_t)gm * C_ + col] = f;
    out[(size_t)gm * OC + col]   = fmaxf(f, 0.f); // relu(feats) half of output
  }
}

// ---------------------------------------------------------------------------
// Kernel 2: e_s[row] = feats[row,:] . a_self ; e_n likewise. One wave per row,
// 4 floats/lane, wave32 shuffle reduction.
// ---------------------------------------------------------------------------
__global__ void __launch_bounds__(256)
gat_logit_dots(const float* __restrict__ feats, const float* __restrict__ a_self,
               const float* __restrict__ a_neigh, float* __restrict__ e_s,
               float* __restrict__ e_n) {
  const int tid  = threadIdx.x;
  const int lane = tid & 31;
  const int row  = blockIdx.x * 8 + (tid >> 5);
  const float4 f  = *(const float4*)(feats + (size_t)row * C_ + lane * 4);
  const float4 as = *(const float4*)(a_self + lane * 4);
  const float4 an = *(const float4*)(a_neigh + lane * 4);
  float s = f.x * as.x + f.y * as.y + f.z * as.z + f.w * as.w;
  float n = f.x * an.x + f.y * an.y + f.z * an.z + f.w * an.w;
#pragma unroll
  for (int off = 16; off; off >>= 1) {
    s += __shfl_down(s, off, 32);
    n += __shfl_down(n, off, 32);
  }
  if (lane == 0) { e_s[row] = s; e_n[row] = n; }
}

// ---------------------------------------------------------------------------
// Kernel 3: fused masked-softmax attention + h = alpha @ feats + b, relu-store.
// Block = 4 waves, all on the SAME batch; each wave owns a 16-row i-tile.
//   pass 1: online softmax stats (2 lanes/row, merged via shfl_xor 16).
//   pass 2: 8 chunks of 128 neighbor rows. Each wave TDM-loads its 32-row
//           quarter of feats into LDS (tensor_load_to_lds, TENSORcnt), then
//           all waves run alpha(16x4) x featsLDS(4x16) WMMAs from LDS.
// Note: relu(max_i A[b,i,j]*h[b,j,c]) == relu(h) given binary A w/ self-loops.
// ---------------------------------------------------------------------------
__global__ void __launch_bounds__(128)
gat_attn_gemm(const float* __restrict__ feats, const float* __restrict__ A,
              const float* __restrict__ e_s, const float* __restrict__ e_n,
              const float* __restrict__ bias, float* __restrict__ out) {
  __shared__ float lds_feats[JT * C_];            // 64 KB of the 320 KB WGP LDS

  const int tid  = threadIdx.x;
  const int wave = tid >> 5;
  const int lane = tid & 31;
  const int b    = blockIdx.x >> 4;               // batch (blocks 0-15 -> b=0)
  const int it   = (blockIdx.x & 15) * 4 + wave;  // i-tile 0..63
  const int i0   = it * 16;
  const int r    = lane & 15;
  const int half = lane >> 4;

  const int   gi   = b * N_ + i0 + r;             // this lane's attention row
  const float esr  = e_s[gi];
  const float* Arow = A + (size_t)gi * N_;
  const float* enb  = e_n + b * N_;

  // ---- pass 1: online softmax (lane covers j = 4t + 2*half + {0,1}) ----
  float m = -INFINITY, s = 0.f;
  for (int t = 0; t < N_ / 4; ++t) {
    if ((t & 15) == 0) __builtin_prefetch(Arow + 4 * t + 256, 0, 0);
    const int j = 4 * t + half * 2;
    float v0 = esr + enb[j];
    float v1 = esr + enb[j + 1];
    v0 = v0 > 0.f ? v0 : 0.2f * v0;               // leaky_relu(0.2)
    v1 = v1 > 0.f ? v1 : 0.2f * v1;
    v0 += -1e10f * (1.0f - Arow[j]);              // mask non-edges
    v1 += -1e10f * (1.0f - Arow[j + 1]);
    const float mn = fmaxf(m, fmaxf(v0, v1));
    s = s * __expf(m - mn) + __expf(v0 - mn) + __expf(v1 - mn);
    m = mn;
  }
  const float mo = __shfl_xor(m, 16, 32);         // merge the two row-halves
  const float so = __shfl_xor(s, 16, 32);
  const float M  = fmaxf(m, mo);
  const float S  = s * __expf(m - M) + so * __expf(mo - M);
  const float rinv = 1.0f / S;

  // ---- pass 2: h tile via f32 WMMA, B-operands staged in LDS ----
  v8f acc[8] = {};
  for (int j0 = 0; j0 < N_; j0 += JT) {
#if GAT_HAVE_TDM
    // Each wave DMAs its 32-row quarter of this feats chunk into LDS.
    {
      const unsigned int qrows = JT / 4;          // 32 rows per wave
      const unsigned int lds_off =
          (unsigned int)(unsigned long long)(void*)lds_feats +
          (unsigned int)wave * qrows * C_ * sizeof(float);
      const unsigned long long gaddr = (unsigned long long)(const void*)(
          feats + (size_t)(b * N_ + j0 + wave * qrows) * C_);
      tdm_load_rows(lds_off, gaddr, qrows);
      __builtin_amdgcn_s_wait_tensorcnt(0);
    }
#else
    // Fallback: cooperative register staging (global -> VGPR -> LDS).
    for (int u = tid; u < JT * C_ / 4; u += 128) {
      const float4 v = *(const float4*)(feats + (size_t)(b * N_ + j0) * C_ + u * 4);
      *(float4*)&lds_feats[u * 4] = v;
    }
#endif
    __syncthreads();                              // chunk visible to all waves

    for (int t = 0; t < JT / 4; ++t) {
      const int jr = 4 * t + half * 2;            // row within the LDS chunk
      const int j  = j0 + jr;                     // lane's K slots (A & B agree)
      float v0 = esr + enb[j];
      float v1 = esr + enb[j + 1];
      v0 = v0 > 0.f ? v0 : 0.2f * v0;
      v1 = v1 > 0.f ? v1 : 0.2f * v1;
      v0 += -1e10f * (1.0f - Arow[j]);
      v1 += -1e10f * (1.0f - Arow[j + 1]);
      v2f a; a.x = __expf(v0 - M) * rinv;         // alpha values
             a.y = __expf(v1 - M) * rinv;
      const float* f0 = &lds_feats[jr * C_];
      const float* f1 = f0 + C_;
#pragma unroll
      for (int ct = 0; ct < 8; ++ct) {            // 8 column tiles of C=128
        v2f bb; bb.x = f0[ct * 16 + r]; bb.y = f1[ct * 16 + r];
        acc[ct] = wmma_f32_k4(a, bb, acc[ct]);
      }
    }
    __syncthreads();                              // before chunk overwrite
  }

#pragma unroll
  for (int ct = 0; ct < 8; ++ct) {
    const int c  = ct * 16 + r;
    const float bc = bias[c];
#pragma unroll
    for (int v = 0; v < 8; ++v) {
      const int gm = b * N_ + i0 + v + 8 * half;
      const float h = acc[ct][v] + bc;
      out[(size_t)gm * OC + C_ + c] = fmaxf(h, 0.f);  // relu(agg) == relu(h)
    }
  }
}

// ---------------------------------------------------------------------------
extern "C" void kernel_launch(void* const* d_in, const int* in_sizes, int n_in,
                              void* d_out, int out_size, void* d_ws,
                              size_t ws_size, hipStream_t stream) {
  const float* X       = (const float*)d_in[0];   // [B,N,F]
  const float* A       = (const float*)d_in[1];   // [B,N,N]
  const float* W       = (const float*)d_in[2];   // [F,C]
  const float* bias    = (const float*)d_in[3];   // [C]
  const float* a_self  = (const float*)d_in[4];   // [C,1]
  const float* a_neigh = (const float*)d_in[5];   // [C,1]
  float* out = (float*)d_out;                     // [B,N,2C]

  float* feats = (float*)d_ws;                    // ROWS*C_ floats (1 MB)
  float* e_s   = feats + (size_t)ROWS * C_;       // ROWS floats
  float* e_n   = e_s + ROWS;                      // ROWS floats

  // 1) feats = X@W (+ relu half of output).  1024 wmma-tiles, 4 waves/block.
  gat_feats_gemm<<<256, 128, 0, stream>>>(X, W, feats, out);
  // 2) per-row logits e_s, e_n.
  gat_logit_dots<<<ROWS / 8, 256, 0, stream>>>(feats, a_self, a_neigh, e_s, e_n);
  // 3) fused softmax + TDM-staged alpha@feats + bias + relu half of output.
  gat_attn_gemm<<<32, 128, 0, stream>>>(feats, A, e_s, e_n, bias, out);
}